// DiscreteHMM_15436112462079
// MI455X (gfx1250) — compile-verified
//
#include <hip/hip_runtime.h>
#include <hip/hip_bf16.h>

typedef __attribute__((ext_vector_type(2))) float v2f;
typedef __attribute__((ext_vector_type(4))) float v4f;
typedef __attribute__((ext_vector_type(8))) float v8f;

constexpr int S = 512;   // NUM_STATES
constexpr int O = 1024;  // NUM_OBS
constexpr int B = 64;    // BATCH
constexpr int T = 512;   // timesteps

// ---------------------------------------------------------------------------
// Generic row softmax: mode 1 -> out = x - logZ (normalized log-probs).
// One block (256 threads) per row.
// ---------------------------------------------------------------------------
__global__ void __launch_bounds__(256) row_softmax_kernel(
    const float* __restrict__ in, float* __restrict__ out, int ncol, int log_mode) {
  const int r = blockIdx.x;
  const float* row = in + (size_t)r * ncol;
  float* orow = out + (size_t)r * ncol;
  __shared__ float red[256];

  float lm = -3.4e38f;
  for (int j = threadIdx.x; j < ncol; j += 256) lm = fmaxf(lm, row[j]);
  red[threadIdx.x] = lm;
  __syncthreads();
  for (int s = 128; s > 0; s >>= 1) {
    if (threadIdx.x < s) red[threadIdx.x] = fmaxf(red[threadIdx.x], red[threadIdx.x + s]);
    __syncthreads();
  }
  const float m = red[0];
  __syncthreads();

  float ls = 0.f;
  for (int j = threadIdx.x; j < ncol; j += 256) ls += __expf(row[j] - m);
  red[threadIdx.x] = ls;
  __syncthreads();
  for (int s = 128; s > 0; s >>= 1) {
    if (threadIdx.x < s) red[threadIdx.x] += red[threadIdx.x + s];
    __syncthreads();
  }
  const float sum = red[0];

  if (log_mode) {
    const float lz = m + __logf(sum);
    for (int j = threadIdx.x; j < ncol; j += 256) orow[j] = row[j] - lz;
  } else {
    const float inv = 1.0f / sum;
    for (int j = threadIdx.x; j < ncol; j += 256) orow[j] = __expf(row[j] - m) * inv;
  }
}

// ---------------------------------------------------------------------------
// Row softmax of log_A with output in WMMA-B packed layout:
//   expAp[((nb*64 + ktpair)*32 + lane)*4 + sub*2 + q] = exp(lA[i, j])
//   where i = 8*ktpair + 4*sub + 2*hi + q, lane = hi*16 + (j&15), nb = j>>4.
// Each lane's 16B chunk is the B fragment for 2 consecutive k-steps, so the
// step kernel fetches B with one global_load_b128 per 2 WMMAs (512B/wave,
// fully coalesced).
// ---------------------------------------------------------------------------
__global__ void __launch_bounds__(256) row_softmax_packA_kernel(
    const float* __restrict__ in, float* __restrict__ outp) {
  const int i = blockIdx.x;  // source state (row of A)
  const float* row = in + (size_t)i * S;
  __shared__ float red[256];

  float lm = fmaxf(row[threadIdx.x], row[threadIdx.x + 256]);
  red[threadIdx.x] = lm;
  __syncthreads();
  for (int s = 128; s > 0; s >>= 1) {
    if (threadIdx.x < s) red[threadIdx.x] = fmaxf(red[threadIdx.x], red[threadIdx.x + s]);
    __syncthreads();
  }
  const float m = red[0];
  __syncthreads();

  float ls = __expf(row[threadIdx.x] - m) + __expf(row[threadIdx.x + 256] - m);
  red[threadIdx.x] = ls;
  __syncthreads();
  for (int s = 128; s > 0; s >>= 1) {
    if (threadIdx.x < s) red[threadIdx.x] += red[threadIdx.x + s];
    __syncthreads();
  }
  const float inv = 1.0f / red[0];

  const int ktpair = i >> 3;
  const int subq = (((i >> 2) & 1) << 1) | (i & 1);
  const int hi16 = ((i >> 1) & 1) << 4;
  for (int j = threadIdx.x; j < S; j += 256) {
    const int nb = j >> 4;
    const int lane = hi16 | (j & 15);
    outp[((((size_t)nb * 64 + ktpair) * 32 + lane) << 2) + subq] = __expf(row[j] - m) * inv;
  }
}

// ---------------------------------------------------------------------------
// alpha0[b,j] = lpi_norm[j] + lBn[j, obs[b,0]]
// ---------------------------------------------------------------------------
__global__ void __launch_bounds__(512) init_alpha_kernel(
    const float* __restrict__ lpin, const float* __restrict__ lBn,
    const int* __restrict__ obs, float* __restrict__ alpha0) {
  const int b = blockIdx.x;
  const int j = threadIdx.x;
  const int o = obs[b * T + 0];
  alpha0[(size_t)b * S + j] = lpin[j] + lBn[(size_t)j * O + o];
}

// ---------------------------------------------------------------------------
// One forward-recursion step via FP32 WMMA:
//   m[b] = max_i alpha_in[b,i];  Ea = exp(alpha_in - m)
//   P    = Ea (16x512) x expA (512x512), one 16x16 tile per wave,
//          K-loop: 64 x { ds_load_b128 (A), global_load_b128 (B), 2x wmma }
//   alpha_out[b,j] = m[b] + log(P[b,j]) + lBn[j, obs[b,t]]
// Grid: (B/16, S/64), 128 threads (4 waves).
// EaP LDS layout mirrors expAp: [ktpair][lane][4]; lane-consecutive 16B
// chunks -> conflict-free ds_load_b128.
// ---------------------------------------------------------------------------
__global__ void __launch_bounds__(128) hmm_step_kernel(
    const float* __restrict__ alpha_in, float* __restrict__ alpha_out,
    const float* __restrict__ expAp, const float* __restrict__ lBn,
    const int* __restrict__ obs, int t) {
  __shared__ __align__(16) float EaP[64 * 32 * 4];  // 32KB packed A fragments
  __shared__ float mrow[16];
  __shared__ float red[128];

  const int tid = threadIdx.x;
  const int b0 = blockIdx.x * 16;

  // ---- per-row max over 512 states: 8 chunks of 64 per row ----
  {
    const int r = tid & 15, ch = tid >> 4;
    const float* arow = alpha_in + (size_t)(b0 + r) * S + ch * 64;
    float pm = -3.4e38f;
#pragma unroll 8
    for (int k = 0; k < 64; ++k) pm = fmaxf(pm, arow[k]);
    red[tid] = pm;
  }
  __syncthreads();
  if (tid < 16) {
    float mm = red[tid];
#pragma unroll
    for (int c = 1; c < 8; ++c) mm = fmaxf(mm, red[tid + (c << 4)]);
    mrow[tid] = mm;
  }
  __syncthreads();

  // ---- Ea = exp(alpha - m), written directly in packed WMMA-A layout ----
  for (int idx = tid; idx < 16 * S; idx += 128) {
    const int r = idx >> 9, k = idx & 511;
    const float e = __expf(alpha_in[(size_t)(b0 + r) * S + k] - mrow[r]);
    // didx = ktpair*128 + hi*64 + r*4 + sub*2 + q   (K index k = 8kt'+4sub+2hi+q)
    const int didx = ((k >> 3) << 7) | (((k >> 1) & 1) << 6) | (r << 2) |
                     (((k >> 2) & 1) << 1) | (k & 1);
    EaP[didx] = e;
  }
  __syncthreads();

  // ---- 16x16 tile GEMM: 64 x (2 x V_WMMA_F32_16X16X4_F32) ----
  const int wave = tid >> 5, lane = tid & 31;
  const int nb = blockIdx.y * 4 + wave;  // 16-wide N tile index (0..31)
  const v4f* __restrict__ apv = (const v4f*)EaP + lane;
  const v4f* __restrict__ bpv = (const v4f*)expAp + (size_t)nb * 64 * 32 + lane;

  v8f c = {};
#pragma unroll 8
  for (int kp = 0; kp < 64; ++kp) {
    const v4f av = apv[kp * 32];
    const v4f bv = bpv[kp * 32];
    v2f a0; a0.x = av.x; a0.y = av.y;
    v2f a1; a1.x = av.z; a1.y = av.w;
    v2f b0; b0.x = bv.x; b0.y = bv.y;
    v2f b1; b1.x = bv.z; b1.y = bv.w;
    c = __builtin_amdgcn_wmma_f32_16x16x4_f32(false, a0, false, b0, (short)0, c, false, false);
    c = __builtin_amdgcn_wmma_f32_16x16x4_f32(false, a1, false, b1, (short)0, c, false, false);
  }

  // ---- epilogue: C/D layout VGPR v -> rows v (lanes 0-15) / v+8 (lanes 16-31) ----
  const int lm16 = lane & 15, hi = lane >> 4;
  const int j = nb * 16 + lm16;
#pragma unroll
  for (int v = 0; v < 8; ++v) {
    const int rr = v + (hi << 3);
    const int br = b0 + rr;
    const int o = obs[br * T + t];
    alpha_out[(size_t)br * S + j] = mrow[rr] + __logf(c[v]) + lBn[(size_t)j * O + o];
  }
}

// ---------------------------------------------------------------------------
// ll[b] = logsumexp_j alpha_T[b, :]   (one 256-thread block per batch row)
// ---------------------------------------------------------------------------
__global__ void __launch_bounds__(256) row_lse_kernel(
    const float* __restrict__ a, float* __restrict__ ll) {
  const int b = blockIdx.x;
  const float* row = a + (size_t)b * S;
  __shared__ float red[256];

  float lm = fmaxf(row[threadIdx.x], row[threadIdx.x + 256]);
  red[threadIdx.x] = lm;
  __syncthreads();
  for (int s = 128; s > 0; s >>= 1) {
    if (threadIdx.x < s) red[threadIdx.x] = fmaxf(red[threadIdx.x], red[threadIdx.x + s]);
    __syncthreads();
  }
  const float m = red[0];
  __syncthreads();

  float ls = __expf(row[threadIdx.x] - m) + __expf(row[threadIdx.x + 256] - m);
  red[threadIdx.x] = ls;
  __syncthreads();
  for (int s = 128; s > 0; s >>= 1) {
    if (threadIdx.x < s) red[threadIdx.x] += red[threadIdx.x + s];
    __syncthreads();
  }
  if (threadIdx.x == 0) ll[b] = m + __logf(red[0]);
}

__global__ void __launch_bounds__(64) sum64_kernel(const float* __restrict__ ll,
                                                   float* __restrict__ out) {
  __shared__ float red[64];
  red[threadIdx.x] = ll[threadIdx.x];
  __syncthreads();
  for (int s = 32; s > 0; s >>= 1) {
    if (threadIdx.x < s) red[threadIdx.x] += red[threadIdx.x + s];
    __syncthreads();
  }
  if (threadIdx.x == 0) out[0] = red[0];
}

// ---------------------------------------------------------------------------
// Workspace layout (bytes, 256-aligned):
//   expAp : [0, 1MB)           packed exp(log_softmax(log_A)) B-fragments
//   lBn   : [1MB, 3MB)         log_softmax(log_B, axis=1)
//   lpin  : [3MB, +2KB)        log_softmax(log_pi)
//   a0,a1 : 2 x 128KB          ping-pong alpha
//   ll    : 256B               per-batch log-likelihood
// ---------------------------------------------------------------------------
extern "C" void kernel_launch(void* const* d_in, const int* in_sizes, int n_in,
                              void* d_out, int out_size, void* d_ws, size_t ws_size,
                              hipStream_t stream) {
  const int* obs = (const int*)d_in[0];         // (B, T) int32
  const float* log_pi = (const float*)d_in[1];  // (S,)
  const float* log_A = (const float*)d_in[2];   // (S, S)
  const float* log_B = (const float*)d_in[3];   // (S, O)

  char* ws = (char*)d_ws;
  float* expAp = (float*)(ws);
  float* lBn   = (float*)(ws + 1048576);
  float* lpin  = (float*)(ws + 3145728);
  float* a0    = (float*)(ws + 3147776);
  float* a1    = (float*)(ws + 3278848);
  float* ll    = (float*)(ws + 3409920);

  // Parameter normalization (once per launch).
  row_softmax_packA_kernel<<<S, 256, 0, stream>>>(log_A, expAp);
  row_softmax_kernel<<<S, 256, 0, stream>>>(log_B, lBn, O, /*log_mode=*/1);
  row_softmax_kernel<<<1, 256, 0, stream>>>(log_pi, lpin, S, /*log_mode=*/1);

  init_alpha_kernel<<<B, S, 0, stream>>>(lpin, lBn, obs, a0);

  float* bufs[2] = {a0, a1};
  for (int t = 1; t < T; ++t) {
    float* ain = bufs[(t + 1) & 1];
    float* aout = bufs[t & 1];
    hmm_step_kernel<<<dim3(B / 16, S / 64), 128, 0, stream>>>(ain, aout, expAp, lBn, obs, t);
  }

  row_lse_kernel<<<B, 256, 0, stream>>>(bufs[(T - 1) & 1], ll);
  sum64_kernel<<<1, 64, 0, stream>>>(ll, (float*)d_out);
}